// SimpleGGNN_16063177687560
// MI455X (gfx1250) — compile-verified
//
#include <hip/hip_runtime.h>

typedef float v2f __attribute__((ext_vector_type(2)));
typedef float v8f __attribute__((ext_vector_type(8)));

#define HID 128
#define LSTR 132          // LDS row stride (floats), padded to avoid bank conflicts
#define NT_MAX 16         // max edge types staged in LDS
#define MTILE 32          // nodes per block (two 16-row WMMA tiles -> 2x B reuse)

// ---------------------------------------------------------------- zero scratch
__global__ __launch_bounds__(256) void ggnn_zero(float* __restrict__ buf, long n) {
  long i = (long)blockIdx.x * 256 + threadIdx.x;
  if (i < n) buf[i] = 0.0f;   // bit-zero also valid for the int count region
}

// ---------------------------------------------------- edge scatter (1 wave/edge)
// agg[dst] += h[src]   (128 floats, 4 per lane)
// cnt[dst][clip(type)] += 1
__global__ __launch_bounds__(256) void ggnn_scatter(
    const float* __restrict__ h, const int* __restrict__ ei,
    const int* __restrict__ et, float* __restrict__ agg,
    int* __restrict__ cnt, int E, int T) {
  int e = blockIdx.x * 8 + (threadIdx.x >> 5);
  if (e >= E) return;
  int lane = threadIdx.x & 31;
  int src = ei[e];
  int dst = ei[E + e];
  if (lane == 0) {
    int t = et[e];
    t = t < 0 ? 0 : (t > T - 1 ? T - 1 : t);
    atomicAdd(&cnt[(long)dst * T + t], 1);
  }
  const float4* hs = (const float4*)(h + (long)src * HID);
  float4 v = hs[lane];
  float* a = agg + (long)dst * HID + lane * 4;
  atomicAdd(a + 0, v.x);
  atomicAdd(a + 1, v.y);
  atomicAdd(a + 2, v.z);
  atomicAdd(a + 3, v.w);
}

// ------------------------------------------- fused per-node GEMMs + GRU (WMMA)
// block = 32 nodes, 256 threads = 8 waves; wave w owns output columns [16w,16w+16)
// Each wave keeps TWO M-tiles (rows 0-15, 16-31) so every B fragment feeds 2 WMMAs.
__global__ __launch_bounds__(256) void ggnn_node(
    const float* __restrict__ h, const float* __restrict__ agg,
    const int* __restrict__ cnt,
    const float* __restrict__ Wmsg, const float* __restrict__ bmsg,
    const float* __restrict__ emb,
    const float* __restrict__ Wih, const float* __restrict__ Whh,
    const float* __restrict__ bih, const float* __restrict__ bhh,
    float* __restrict__ out, int N, int T) {
  __shared__ float s_agg[MTILE * LSTR];
  __shared__ float s_h[MTILE * LSTR];
  __shared__ float s_x[MTILE * LSTR];
  __shared__ float s_cnt[MTILE * NT_MAX];

  const int tid   = threadIdx.x;
  const int wave  = tid >> 5;
  const int lane  = tid & 31;
  const int half  = lane >> 4;        // which 16-lane half
  const int nloc  = lane & 15;        // N (or M) index within tile
  const int node0 = blockIdx.x * MTILE;

  // ---- stage agg / h tiles: thread covers row tid/8, 16 consecutive cols
  {
    int row  = tid >> 3;              // 0..31
    int col  = (tid & 7) * 16;        // 0,16,...,112
    int node = node0 + row;
    float4 av[4], hv[4];
    #pragma unroll
    for (int i = 0; i < 4; ++i) {
      av[i] = make_float4(0.f, 0.f, 0.f, 0.f);
      hv[i] = av[i];
    }
    if (node < N) {
      const float4* pa = (const float4*)(agg + (long)node * HID + col);
      const float4* ph = (const float4*)(h   + (long)node * HID + col);
      #pragma unroll
      for (int i = 0; i < 4; ++i) { av[i] = pa[i]; hv[i] = ph[i]; }
    }
    float4* da = (float4*)&s_agg[row * LSTR + col];
    float4* dh = (float4*)&s_h  [row * LSTR + col];
    #pragma unroll
    for (int i = 0; i < 4; ++i) { da[i] = av[i]; dh[i] = hv[i]; }
  }
  // ---- stage per-(node,type) counts as floats
  for (int i = tid; i < MTILE * T; i += 256) {
    int row = i / T, t = i - row * T;
    int node = node0 + row;
    s_cnt[row * NT_MAX + t] =
        (node < N) ? (float)cnt[(long)node * T + t] : 0.0f;
  }
  __syncthreads();

  // =================== phase 1: x = agg @ Wmsg^T + deg*bmsg + counts @ emb ====
  {
    v8f acc[2];
    acc[0] = (v8f){0.f,0.f,0.f,0.f,0.f,0.f,0.f,0.f};
    acc[1] = acc[0];
    const int n = wave * 16 + nloc;                 // output column
    const float* pA0 = &s_agg[nloc * LSTR + 2 * half];
    const float* pA1 = &s_agg[(16 + nloc) * LSTR + 2 * half];
    const float* pB  = Wmsg + (long)n * HID + 2 * half;  // B[k][n] = Wmsg[n][k]
    #pragma unroll 8
    for (int kb = 0; kb < HID; kb += 4) {
      v2f b  = *(const v2f*)(pB + kb);
      v2f a0 = *(const v2f*)(pA0 + kb);
      v2f a1 = *(const v2f*)(pA1 + kb);
      acc[0] = __builtin_amdgcn_wmma_f32_16x16x4_f32(
          false, a0, false, b, (short)0, acc[0], false, false);
      acc[1] = __builtin_amdgcn_wmma_f32_16x16x4_f32(
          false, a1, false, b, (short)0, acc[1], false, false);
    }
    // epilogue: bias * degree + counts @ edge_embed (tiny T-term dot per element)
    float bm = bmsg[n];
    float eb[NT_MAX];
    for (int t = 0; t < T; ++t) eb[t] = emb[(long)t * HID + n];
    #pragma unroll
    for (int mt = 0; mt < 2; ++mt) {
      #pragma unroll
      for (int v = 0; v < 8; ++v) {
        int m = mt * 16 + v + 8 * half;
        float deg = 0.f, add = 0.f;
        for (int t = 0; t < T; ++t) {
          float c = s_cnt[m * NT_MAX + t];
          deg += c;
          add += c * eb[t];
        }
        s_x[m * LSTR + n] = acc[mt][v] + bm * deg + add;
      }
    }
  }
  __syncthreads();

  // ====== phase 2: gi = x @ Wih^T + bih ; gh = h @ Whh^T + bhh ; GRU gates ====
  {
    v8f air[2], aiz[2], ain[2], ahr[2], ahz[2], ahn[2];
    #pragma unroll
    for (int mt = 0; mt < 2; ++mt) {
      air[mt] = (v8f){0.f,0.f,0.f,0.f,0.f,0.f,0.f,0.f};
      aiz[mt] = air[mt]; ain[mt] = air[mt];
      ahr[mt] = air[mt]; ahz[mt] = air[mt]; ahn[mt] = air[mt];
    }
    const int n = wave * 16 + nloc;
    const float* pAx[2] = { &s_x[nloc * LSTR + 2 * half],
                            &s_x[(16 + nloc) * LSTR + 2 * half] };
    const float* pAh[2] = { &s_h[nloc * LSTR + 2 * half],
                            &s_h[(16 + nloc) * LSTR + 2 * half] };
    const long ko = 2 * half;
    const float* pBir = Wih + (long)(0 * HID + n) * HID + ko;
    const float* pBiz = Wih + (long)(1 * HID + n) * HID + ko;
    const float* pBin = Wih + (long)(2 * HID + n) * HID + ko;
    const float* pBhr = Whh + (long)(0 * HID + n) * HID + ko;
    const float* pBhz = Whh + (long)(1 * HID + n) * HID + ko;
    const float* pBhn = Whh + (long)(2 * HID + n) * HID + ko;
    #pragma unroll 2
    for (int kb = 0; kb < HID; kb += 4) {
      v2f bir = *(const v2f*)(pBir + kb);
      v2f biz = *(const v2f*)(pBiz + kb);
      v2f bin = *(const v2f*)(pBin + kb);
      v2f bhr = *(const v2f*)(pBhr + kb);
      v2f bhz = *(const v2f*)(pBhz + kb);
      v2f bhn = *(const v2f*)(pBhn + kb);
      #pragma unroll
      for (int mt = 0; mt < 2; ++mt) {
        v2f ax = *(const v2f*)(pAx[mt] + kb);
        v2f ah = *(const v2f*)(pAh[mt] + kb);
        air[mt] = __builtin_amdgcn_wmma_f32_16x16x4_f32(
            false, ax, false, bir, (short)0, air[mt], false, false);
        aiz[mt] = __builtin_amdgcn_wmma_f32_16x16x4_f32(
            false, ax, false, biz, (short)0, aiz[mt], false, false);
        ain[mt] = __builtin_amdgcn_wmma_f32_16x16x4_f32(
            false, ax, false, bin, (short)0, ain[mt], false, false);
        ahr[mt] = __builtin_amdgcn_wmma_f32_16x16x4_f32(
            false, ah, false, bhr, (short)0, ahr[mt], false, false);
        ahz[mt] = __builtin_amdgcn_wmma_f32_16x16x4_f32(
            false, ah, false, bhz, (short)0, ahz[mt], false, false);
        ahn[mt] = __builtin_amdgcn_wmma_f32_16x16x4_f32(
            false, ah, false, bhn, (short)0, ahn[mt], false, false);
      }
    }
    float bir_s = bih[n],           bhr_s = bhh[n];
    float biz_s = bih[HID + n],     bhz_s = bhh[HID + n];
    float bin_s = bih[2 * HID + n], bhn_s = bhh[2 * HID + n];
    #pragma unroll
    for (int mt = 0; mt < 2; ++mt) {
      #pragma unroll
      for (int v = 0; v < 8; ++v) {
        int m = mt * 16 + v + 8 * half;
        float r  = 1.f / (1.f + __expf(-(air[mt][v] + bir_s + ahr[mt][v] + bhr_s)));
        float z  = 1.f / (1.f + __expf(-(aiz[mt][v] + biz_s + ahz[mt][v] + bhz_s)));
        float nn = tanhf(ain[mt][v] + bin_s + r * (ahn[mt][v] + bhn_s));
        float hp = s_h[m * LSTR + n];
        int node = node0 + m;
        if (node < N) out[(long)node * HID + n] = (1.f - z) * nn + z * hp;
      }
    }
  }
}

// ----------------------------------------------------------------------- host
extern "C" void kernel_launch(void* const* d_in, const int* in_sizes, int n_in,
                              void* d_out, int out_size, void* d_ws, size_t ws_size,
                              hipStream_t stream) {
  const float* h    = (const float*)d_in[0];
  const int*   ei   = (const int*)d_in[1];
  const int*   et   = (const int*)d_in[2];
  const float* Wmsg = (const float*)d_in[4];
  const float* bmsg = (const float*)d_in[5];
  const float* emb  = (const float*)d_in[6];
  const float* Wih  = (const float*)d_in[7];
  const float* Whh  = (const float*)d_in[8];
  const float* bih  = (const float*)d_in[9];
  const float* bhh  = (const float*)d_in[10];

  const int N = in_sizes[0] / HID;      // 50000
  const int E = in_sizes[2];            // 600000
  const int T = in_sizes[6] / HID;      // 9 edge-type embeddings

  float* agg = (float*)d_ws;                                    // N*HID f32
  int*   cnt = (int*)((char*)d_ws + (size_t)N * HID * sizeof(float)); // N*T i32

  long zwords = (long)N * HID + (long)N * T;
  ggnn_zero<<<(int)((zwords + 255) / 256), 256, 0, stream>>>((float*)d_ws, zwords);
  ggnn_scatter<<<(E + 7) / 8, 256, 0, stream>>>(h, ei, et, agg, cnt, E, T);
  ggnn_node<<<(N + MTILE - 1) / MTILE, 256, 0, stream>>>(h, agg, cnt, Wmsg, bmsg,
                                                         emb, Wih, Whh, bih, bhh,
                                                         (float*)d_out, N, T);
}